// Group_927712936450
// MI455X (gfx1250) — compile-verified
//
#include <hip/hip_runtime.h>
#include <hip/hip_bf16.h>

// Problem constants (from reference): xyz [32, 8192, 3] f32
#define BATCH   32
#define NPTS    8192
#define NGROUP  512     // NUM_GROUP
#define KNEIGH  32      // GROUP_SIZE
#define TILE    16      // centers per block in group kernel
#define CHUNK   512     // points per LDS chunk
#define NCHUNK  (NPTS / CHUNK)          // 16
#define TPC     (CHUNK / 16)            // 32 point-tiles per chunk
#define FPS_T   1024
#define PPT     (NPTS / FPS_T)          // 8 points per thread in FPS
#define BIGF    3.0e38f
#define ALLDONE ((1 << TILE) - 1)

typedef __attribute__((ext_vector_type(2))) float v2f;
typedef __attribute__((ext_vector_type(4))) float v4f;
typedef __attribute__((ext_vector_type(8))) float v8f;
typedef __attribute__((ext_vector_type(4))) int   v4i;

#define HAS_ASYNC_LDS __has_builtin(__builtin_amdgcn_global_load_async_to_lds_b128)

// ---------------------------------------------------------------------------
// Kernel 1: furthest point sampling. One block per batch, dists in LDS,
// each thread's 8 points cached in registers (FPS is 512 *serial* rounds,
// so the inner loop must be pure VALU/LDS).
// ---------------------------------------------------------------------------
__global__ __launch_bounds__(FPS_T)
void fps_kernel(const float* __restrict__ xyz, float* __restrict__ centers /*[B][G][3]*/) {
    __shared__ float dists[NPTS];     // 32 KB
    __shared__ float redV[32];
    __shared__ int   redI[32];
    __shared__ int   s_far;

    const int b   = blockIdx.x;
    const int tid = threadIdx.x;
    const float* xb = xyz + (size_t)b * NPTS * 3;

    float px[PPT], py[PPT], pz[PPT];
    #pragma unroll
    for (int k = 0; k < PPT; ++k) {
        const int n = tid + k * FPS_T;
        px[k] = xb[n * 3 + 0];
        py[k] = xb[n * 3 + 1];
        pz[k] = xb[n * 3 + 2];
    }

    for (int n = tid; n < NPTS; n += FPS_T) dists[n] = 1e10f;
    if (tid == 0) s_far = 0;
    __syncthreads();

    for (int it = 0; it < NGROUP; ++it) {
        const int far = s_far;
        const float cx = xb[far * 3 + 0];   // broadcast load, L2-resident
        const float cy = xb[far * 3 + 1];
        const float cz = xb[far * 3 + 2];
        if (tid == 0) {
            float* c = centers + ((size_t)b * NGROUP + it) * 3;
            c[0] = cx; c[1] = cy; c[2] = cz;
        }
        float bv = -1.0f; int bi = 0;
        #pragma unroll
        for (int k = 0; k < PPT; ++k) {
            const int n = tid + k * FPS_T;
            const float dx = px[k] - cx;
            const float dy = py[k] - cy;
            const float dz = pz[k] - cz;
            const float d  = dx * dx + dy * dy + dz * dz;
            const float v  = fminf(dists[n], d);
            dists[n] = v;
            if (v > bv) { bv = v; bi = n; }
        }
        #pragma unroll
        for (int off = 16; off > 0; off >>= 1) {
            const float ov = __shfl_xor(bv, off, 32);
            const int   oi = __shfl_xor(bi, off, 32);
            if (ov > bv || (ov == bv && oi < bi)) { bv = ov; bi = oi; }
        }
        const int wave = tid >> 5;
        if ((tid & 31) == 0) { redV[wave] = bv; redI[wave] = bi; }
        __syncthreads();
        if (wave == 0) {
            float v = redV[tid & 31];
            int   i = redI[tid & 31];
            #pragma unroll
            for (int off = 16; off > 0; off >>= 1) {
                const float ov = __shfl_xor(v, off, 32);
                const int   oi = __shfl_xor(i, off, 32);
                if (ov > v || (ov == v && oi < i)) { v = ov; i = oi; }
            }
            if (tid == 0) s_far = i;
        }
        __syncthreads();
    }
}

// ---------------------------------------------------------------------------
// Async staging of one 512-point chunk (6144 B = 384 x b128) into LDS.
// Uses GLOBAL_LOAD_ASYNC_TO_LDS (ASYNCcnt) when available; sync fallback else.
// ---------------------------------------------------------------------------
__device__ __forceinline__ void stage_chunk(const float* __restrict__ src,
                                            float* lds_dst, int tid) {
#if HAS_ASYNC_LDS
    #pragma unroll
    for (int pass = 0; pass < 2; ++pass) {
        const int e = tid + pass * 256;             // 384 x 16B transfers
        if (e < (CHUNK * 3) / 4) {
            __builtin_amdgcn_global_load_async_to_lds_b128(
                (v4i*)(src + e * 4),
                (v4i*)(lds_dst + e * 4),
                0, 0);
        }
    }
#else
    #pragma unroll
    for (int pass = 0; pass < 2; ++pass) {
        const int e = tid + pass * 256;
        if (e < (CHUNK * 3) / 4) {
            const v4f v = *(const v4f*)(src + e * 4);
            *(v4f*)(lds_dst + e * 4) = v;
        }
    }
#endif
}

__device__ __forceinline__ void wait_async_done() {
#if HAS_ASYNC_LDS
  #if __has_builtin(__builtin_amdgcn_s_wait_asynccnt)
    __builtin_amdgcn_s_wait_asynccnt(0);
  #else
    asm volatile("s_wait_asynccnt 0x0" ::: "memory");
  #endif
#endif
}

// ---------------------------------------------------------------------------
// Kernel 2: distances via V_WMMA_F32_16X16X4_F32 + exact running top-32,
// with async double-buffered point staging and early-exit merge rounds.
// Block = 256 threads (8 waves) handles 16 centers of one batch.
// ---------------------------------------------------------------------------
__global__ __launch_bounds__(256)
void group_kernel(const float* __restrict__ xyz,
                  const float* __restrict__ centers,   // [B][G][3] (from d_out)
                  float* __restrict__ out_neigh) {     // [B][G][K][3]
    __shared__ float dchunk[TILE][CHUNK + 1];          // ~32.8 KB, row-padded
    __shared__ __align__(16) float pbuf[2][CHUNK * 3]; // 12 KB point double-buffer
    __shared__ float bestD[TILE][KNEIGH];
    __shared__ int   bestI[TILE][KNEIGH];
    __shared__ float worstV[TILE];
    __shared__ int   worstP[TILE];
    __shared__ float scx[TILE], scy[TILE], scz[TILE], scn[TILE];
    __shared__ int   doneMask;

    const int tileId = blockIdx.x;          // 0..G/TILE-1
    const int b      = blockIdx.y;          // 0..B-1
    const int tid    = threadIdx.x;
    const int lane   = tid & 31;
    const int wave   = tid >> 5;            // 0..7
    const int half   = lane >> 4;           // hi 16 lanes hold M=8..15 rows of D
    const int l16    = lane & 15;

    const float* xb = xyz + (size_t)b * NPTS * 3;
    const float* cb = centers + ((size_t)b * NGROUP + tileId * TILE) * 3;

    // stage chunk 0 while we set up centers/lists
    stage_chunk(xb, &pbuf[0][0], tid);

    if (tid < TILE) {
        const float cx = cb[tid * 3 + 0], cy = cb[tid * 3 + 1], cz = cb[tid * 3 + 2];
        scx[tid] = cx; scy[tid] = cy; scz[tid] = cz;
        scn[tid] = cx * cx + cy * cy + cz * cz;
        worstV[tid] = BIGF; worstP[tid] = 0;
    }
    for (int k = tid; k < TILE * KNEIGH; k += 256) {
        (&bestD[0][0])[k] = BIGF;
        (&bestI[0][0])[k] = 0;
    }
    wait_async_done();
    __syncthreads();

    // A operand: 16x4 f32, lanes 0-15 carry K=0/1 (x,y), lanes 16-31 K=2/3 (z,pad0)
    v2f A;
    if (half == 0) { A.x = -2.0f * scx[l16]; A.y = -2.0f * scy[l16]; }
    else           { A.x = -2.0f * scz[l16]; A.y = 0.0f; }
    float cn[8];
    #pragma unroll
    for (int v = 0; v < 8; ++v) cn[v] = scn[v + 8 * half];

    const int c = tid >> 4;       // center 0..15 (selection role)
    const int s = tid & 15;

    for (int chunk = 0; chunk < NCHUNK; ++chunk) {
        const int cur = chunk & 1;
        const float* pb = &pbuf[cur][0];

        // ---- distance tiles from LDS-staged points: 4 WMMAs per wave ----
        #pragma unroll
        for (int i = 0; i < TPC / 8; ++i) {
            const int t  = wave + 8 * i;          // wave-uniform: EXEC all ones
            const int pl = t * 16 + l16;
            const float ptx = pb[pl * 3 + 0];
            const float pty = pb[pl * 3 + 1];
            const float ptz = pb[pl * 3 + 2];
            v2f Bm;
            if (half == 0) { Bm.x = ptx; Bm.y = pty; }
            else           { Bm.x = ptz; Bm.y = 0.0f; }
            const float xn = ptx * ptx + pty * pty + ptz * ptz;
            v8f C;
            #pragma unroll
            for (int v = 0; v < 8; ++v) C[v] = cn[v] + xn;
            v8f D = __builtin_amdgcn_wmma_f32_16x16x4_f32(
                        false, A, false, Bm, (short)0, C, false, false);
            #pragma unroll
            for (int v = 0; v < 8; ++v)
                dchunk[v + 8 * half][t * 16 + l16] = D[v];
        }
        if (tid == 0) doneMask = 0;

        // overlap: stage next chunk during the selection rounds below
        if (chunk + 1 < NCHUNK)
            stage_chunk(xb + (size_t)(chunk + 1) * CHUNK * 3, &pbuf[cur ^ 1][0], tid);
        __syncthreads();

        // ---- exact merge into running top-32, uniform early exit ----
        for (int r = 0; r < KNEIGH; ++r) {
            const int dm = doneMask;              // read after a barrier: uniform
            if (dm == ALLDONE) break;             // convergent break
            if (!((dm >> c) & 1)) {
                float mv = BIGF; int mi = -1;
                for (int j = s; j < CHUNK; j += 16) {
                    const float d = dchunk[c][j];
                    if (d < mv) { mv = d; mi = j; }
                }
                #pragma unroll
                for (int off = 8; off > 0; off >>= 1) {   // within 16-lane group
                    const float ov = __shfl_xor(mv, off, 32);
                    const int   oi = __shfl_xor(mi, off, 32);
                    if (ov < mv || (ov == mv && oi < mi)) { mv = ov; mi = oi; }
                }
                if (s == 0) {
                    if (mv < worstV[c]) {
                        const int wp = worstP[c];
                        bestD[c][wp] = mv;
                        bestI[c][wp] = chunk * CHUNK + mi;
                        dchunk[c][mi] = BIGF;
                        float wv = -1.0f; int wpos = 0;
                        #pragma unroll
                        for (int k2 = 0; k2 < KNEIGH; ++k2) {
                            const float d = bestD[c][k2];
                            if (d > wv) { wv = d; wpos = k2; }
                        }
                        worstV[c] = wv; worstP[c] = wpos;
                    } else {
                        atomicOr(&doneMask, 1 << c);   // center saturated for chunk
                    }
                }
            }
            __syncthreads();
        }

        // wait for next chunk's async stage + epoch fence (doneMask/dchunk reuse)
        wait_async_done();
        __syncthreads();
    }

    // ---- order neighbors by ascending distance (matches top_k ordering) ----
    if ((tid & 15) == 0) {
        for (int i = 0; i < KNEIGH - 1; ++i) {
            int m = i;
            for (int j = i + 1; j < KNEIGH; ++j) {
                if (bestD[c][j] < bestD[c][m] ||
                    (bestD[c][j] == bestD[c][m] && bestI[c][j] < bestI[c][m])) m = j;
            }
            const float td = bestD[c][m]; bestD[c][m] = bestD[c][i]; bestD[c][i] = td;
            const int   ti = bestI[c][m]; bestI[c][m] = bestI[c][i]; bestI[c][i] = ti;
        }
    }
    __syncthreads();

    // ---- gather + subtract center, write [B][G][K][3] ----
    for (int e = tid; e < TILE * KNEIGH; e += 256) {
        const int cc = e >> 5;          // / KNEIGH
        const int k  = e & (KNEIGH - 1);
        const int idx = bestI[cc][k];
        const int g = tileId * TILE + cc;
        float* o = out_neigh + (((size_t)b * NGROUP + g) * KNEIGH + k) * 3;
        o[0] = xb[idx * 3 + 0] - scx[cc];
        o[1] = xb[idx * 3 + 1] - scy[cc];
        o[2] = xb[idx * 3 + 2] - scz[cc];
    }
}

// ---------------------------------------------------------------------------
extern "C" void kernel_launch(void* const* d_in, const int* in_sizes, int n_in,
                              void* d_out, int out_size, void* d_ws, size_t ws_size,
                              hipStream_t stream) {
    (void)in_sizes; (void)n_in; (void)d_ws; (void)ws_size; (void)out_size;
    const float* xyz = (const float*)d_in[0];
    float* out        = (float*)d_out;
    float* out_neigh  = out;                                       // B*G*K*3 floats
    float* out_center = out + (size_t)BATCH * NGROUP * KNEIGH * 3; // B*G*3 floats

    // FPS writes centers directly into d_out; group kernel reads them back.
    fps_kernel<<<BATCH, FPS_T, 0, stream>>>(xyz, out_center);

    dim3 grid(NGROUP / TILE, BATCH);
    group_kernel<<<grid, 256, 0, stream>>>(xyz, out_center, out_neigh);
}